// SpecialSpmmFinal_17918603559513
// MI455X (gfx1250) — compile-verified
//
#include <hip/hip_runtime.h>

typedef __attribute__((ext_vector_type(2))) float v2f;
typedef __attribute__((ext_vector_type(4))) float v4f;
typedef __attribute__((ext_vector_type(8))) float v8f;

// ---------------- zero-fill helpers ----------------
__global__ void zero_u32_kernel(unsigned* __restrict__ p, long n) {
    long i = (long)blockIdx.x * blockDim.x + threadIdx.x;
    if (i < n) p[i] = 0u;
}

__global__ void zero_f32_kernel(float* __restrict__ p, long n) {
    long i = (long)blockIdx.x * blockDim.x + threadIdx.x;
    if (i < n) p[i] = 0.0f;
}

// ---------------- pass 1: per-node degree count ----------------
__global__ void count_kernel(const int* __restrict__ src, unsigned* __restrict__ counts, long E) {
    long e = (long)blockIdx.x * blockDim.x + threadIdx.x;
    if (e < E) atomicAdd(&counts[src[e]], 1u);
}

// ---------------- pass 2: exclusive scan (single block, in-place) ----------
__global__ __launch_bounds__(1024)
void scan_kernel(unsigned* __restrict__ off, unsigned* __restrict__ cur, long N) {
    __shared__ unsigned s[1024];
    const int t = threadIdx.x;
    unsigned carry = 0u;
    for (long base = 0; base < N; base += 1024) {
        long i = base + t;
        unsigned v = (i < N) ? off[i] : 0u;
        s[t] = v;
        __syncthreads();
        for (int o = 1; o < 1024; o <<= 1) {
            unsigned x = (t >= o) ? s[t - o] : 0u;
            __syncthreads();
            s[t] += x;
            __syncthreads();
        }
        unsigned incl  = s[t];
        unsigned total = s[1023];
        __syncthreads();                  // protect s[] before next iteration's store
        if (i < N) {
            unsigned excl = carry + incl - v;
            off[i] = excl;
            cur[i] = excl;
        }
        carry += total;
    }
    if (t == 0) off[N] = carry;
}

// ---------------- pass 3: build edge permutation (CSR fill) ----------------
__global__ void fill_kernel(const int* __restrict__ src, unsigned* __restrict__ cur,
                            unsigned* __restrict__ perm, long E) {
    long e = (long)blockIdx.x * blockDim.x + threadIdx.x;
    if (e < E) {
        unsigned pos = atomicAdd(&cur[src[e]], 1u);
        perm[pos] = (unsigned)e;
    }
}

// ---------------- pass 4: atomic-free gather-reduce via WMMA ----------------
// One wave32 per node. Sum of K edge-rows == ones(16x4) x B(4x64-chunk).
// A == all-ones makes the result invariant to the exact K<->(vgpr,lane-half)
// mapping, and all 16 rows of D are identical, so element 0 of each
// accumulator at lane l is the column sum for column (l & 15).
// Feature mapping: chunk c holds feature 4*col + c  =>  per-lane B data is one
// contiguous float4 (global_load_b128) and the output store is one b128.

__device__ __forceinline__ void accum4(v2f a, v4f wA, v4f wB,
                                       v8f& a0, v8f& a1, v8f& a2, v8f& a3) {
    v2f b0 = {wA[0], wB[0]};
    v2f b1 = {wA[1], wB[1]};
    v2f b2 = {wA[2], wB[2]};
    v2f b3 = {wA[3], wB[3]};
    a0 = __builtin_amdgcn_wmma_f32_16x16x4_f32(false, a, false, b0, (short)0, a0, false, false);
    a1 = __builtin_amdgcn_wmma_f32_16x16x4_f32(false, a, false, b1, (short)0, a1, false, false);
    a2 = __builtin_amdgcn_wmma_f32_16x16x4_f32(false, a, false, b2, (short)0, a2, false, false);
    a3 = __builtin_amdgcn_wmma_f32_16x16x4_f32(false, a, false, b3, (short)0, a3, false, false);
}

__global__ __launch_bounds__(128)
void reduce_wmma_kernel(const float* __restrict__ W, const unsigned* __restrict__ off,
                        const unsigned* __restrict__ perm, float* __restrict__ out, long N) {
    long node = (long)blockIdx.x * 4 + threadIdx.y;   // wave-uniform
    if (node >= N) return;                            // whole wave exits together

    const int lane = threadIdx.x;        // 0..31
    const int n    = lane & 15;          // feature column within chunk
    const int h    = lane >> 4;          // lane-half selects edge slot

    unsigned start = off[node];
    unsigned end   = off[node + 1];

    v8f acc0 = {}, acc1 = {}, acc2 = {}, acc3 = {};
    v2f a = {1.0f, 1.0f};                // all-ones A matrix (16x4)

    unsigned e = start;

    // Dense main loop: 8 edges per iteration, no masking.
    // 4 independent perm loads -> one wait -> 4 independent B128 loads ->
    // one wait -> 8 WMMAs. Maximizes loads in flight on the streaming path.
    for (; e + 8 <= end; e += 8) {
        unsigned iA0 = perm[e + 0 + h];
        unsigned iB0 = perm[e + 2 + h];
        unsigned iA1 = perm[e + 4 + h];
        unsigned iB1 = perm[e + 6 + h];
        const v4f* pA0 = (const v4f*)(W + (size_t)iA0 * 64 + 4 * n);
        const v4f* pB0 = (const v4f*)(W + (size_t)iB0 * 64 + 4 * n);
        const v4f* pA1 = (const v4f*)(W + (size_t)iA1 * 64 + 4 * n);
        const v4f* pB1 = (const v4f*)(W + (size_t)iB1 * 64 + 4 * n);
        v4f wA0 = __builtin_nontemporal_load(pA0);   // edge_w streamed once: NT
        v4f wB0 = __builtin_nontemporal_load(pB0);
        v4f wA1 = __builtin_nontemporal_load(pA1);
        v4f wB1 = __builtin_nontemporal_load(pB1);
        accum4(a, wA0, wB0, acc0, acc1, acc2, acc3);
        accum4(a, wA1, wB1, acc0, acc1, acc2, acc3);
    }

    // Dense 4-edge block (at most one).
    if (e + 4 <= end) {
        unsigned iA = perm[e + 0 + h];
        unsigned iB = perm[e + 2 + h];
        v4f wA = __builtin_nontemporal_load((const v4f*)(W + (size_t)iA * 64 + 4 * n));
        v4f wB = __builtin_nontemporal_load((const v4f*)(W + (size_t)iB * 64 + 4 * n));
        accum4(a, wA, wB, acc0, acc1, acc2, acc3);
        e += 4;
    }

    // Masked tail (0..3 edges): divergence only touches the loads.
    if (e < end) {
        v4f wA = {0.f, 0.f, 0.f, 0.f};
        v4f wB = {0.f, 0.f, 0.f, 0.f};
        unsigned eA = e + (unsigned)h;
        unsigned eB = e + 2u + (unsigned)h;
        if (eA < end) {
            unsigned id = perm[eA];
            wA = __builtin_nontemporal_load((const v4f*)(W + (size_t)id * 64 + 4 * n));
        }
        if (eB < end) {
            unsigned id = perm[eB];
            wB = __builtin_nontemporal_load((const v4f*)(W + (size_t)id * 64 + 4 * n));
        }
        accum4(a, wA, wB, acc0, acc1, acc2, acc3);
    }

    // Lanes 0-15 write the full 64-feature row as one contiguous b128.
    if (h == 0) {
        v4f r = {acc0[0], acc1[0], acc2[0], acc3[0]};
        __builtin_nontemporal_store(r, (v4f*)(out + (size_t)node * 64 + 4 * n));
    }
}

// ---------------- fallback: direct f32 atomic scatter ----------------
__global__ void scatter_atomic_kernel(const int* __restrict__ src, const float* __restrict__ W,
                                      float* __restrict__ out, long E) {
    long t = (long)blockIdx.x * blockDim.x + threadIdx.x;
    long e = t >> 5;
    int  j = (int)(t & 31);
    if (e >= E) return;
    int idx = src[e];
    float2 v = *(const float2*)(W + e * 64 + 2 * j);
    atomicAdd(out + (size_t)idx * 64 + 2 * j,     v.x);
    atomicAdd(out + (size_t)idx * 64 + 2 * j + 1, v.y);
}

extern "C" void kernel_launch(void* const* d_in, const int* in_sizes, int n_in,
                              void* d_out, int out_size, void* d_ws, size_t ws_size,
                              hipStream_t stream) {
    const int*   edge = (const int*)d_in[0];   // (2, E) int32; row 0 = source idx
    const float* W    = (const float*)d_in[1]; // (E, 64) f32
    float*       out  = (float*)d_out;         // (N, 64) f32

    const long E = (long)in_sizes[0] / 2;
    const long N = (long)out_size / 64;
    const int* src = edge;                     // edge[0]

    const size_t need = (size_t)(2 * N + 1 + E) * sizeof(unsigned);

    if (ws_size >= need) {
        unsigned* off  = (unsigned*)d_ws;      // N+1 (counts -> offsets in place)
        unsigned* cur  = off + (N + 1);        // N    (scan copy, atomic cursors)
        unsigned* perm = cur + N;              // E    (edge ids grouped by node)

        zero_u32_kernel<<<(unsigned)((N + 1 + 255) / 256), 256, 0, stream>>>(off, N + 1);
        count_kernel   <<<(unsigned)((E + 255) / 256),     256, 0, stream>>>(src, off, E);
        scan_kernel    <<<1, 1024, 0, stream>>>(off, cur, N);
        fill_kernel    <<<(unsigned)((E + 255) / 256),     256, 0, stream>>>(src, cur, perm, E);
        reduce_wmma_kernel<<<(unsigned)((N + 3) / 4), dim3(32, 4), 0, stream>>>(W, off, perm, out, N);
    } else {
        // Workspace too small: bandwidth-correct but atomic-bound fallback.
        zero_f32_kernel<<<(unsigned)((N * 64 + 255) / 256), 256, 0, stream>>>(out, N * 64);
        long threads = E * 32;
        scatter_atomic_kernel<<<(unsigned)((threads + 255) / 256), 256, 0, stream>>>(src, W, out, E);
    }
}